// Quantizer_33380485824526
// MI455X (gfx1250) — compile-verified
//
#include <hip/hip_runtime.h>

typedef float v2f __attribute__((ext_vector_type(2)));
typedef float v8f __attribute__((ext_vector_type(8)));

#define N_ROWS   131072          // 32*4096
#define DIM      64
#define NE       512
#define NTILES   (NE / 16)       // 32 code tiles
#define KSTEPS   (DIM / 4)       // 16 wmma k-steps

// d_out flat layout (reference tuple order)
#define OFF_Q    0
#define OFF_DIFF (N_ROWS * DIM)          // 8388608
#define OFF_IND  (OFF_DIFF + 1)
#define OFF_OH   (OFF_IND + N_ROWS)
#define OFF_ES   (OFF_OH + NE)

// d_ws layout: enorm[NE] | embedT[NE*DIM]
#define WS_ENORM 0
#define WS_EMBT  NE

// ---------------------------------------------------------------------------
// Kernel 0: zero accumulator regions of d_out, precompute ||e_n||^2 and the
// transposed codebook embedT[n][k] into workspace.
// ---------------------------------------------------------------------------
__global__ void vq_init(const float* __restrict__ embed,
                        float* __restrict__ out,
                        float* __restrict__ ws) {
    int tid = blockIdx.x * blockDim.x + threadIdx.x;
    if (tid < NE) {
        float s = 0.0f;
        #pragma unroll 8
        for (int k = 0; k < DIM; ++k) {
            float e = embed[k * NE + tid];
            s += e * e;
        }
        ws[WS_ENORM + tid] = s;
        out[OFF_OH + tid] = 0.0f;
    }
    if (tid == 0) out[OFF_DIFF] = 0.0f;
    if (tid < DIM * NE) {
        out[OFF_ES + tid] = 0.0f;
        // transpose: tid = n*DIM + k  ->  embedT[tid] = embed[k*NE + n]
        const int n = tid / DIM;
        const int k = tid - n * DIM;
        ws[WS_EMBT + tid] = embed[k * NE + n];
    }
}

// ---------------------------------------------------------------------------
// Kernel 1: per-wave 16-row tile. WMMA fp32 GEMM (x @ embed) -> argmin ->
// codebook gather + scatter accumulators.
// Block = 128 threads = 4 waves; grid exactly tiles N (EXEC all-1s for WMMA).
// ---------------------------------------------------------------------------
__global__ void __launch_bounds__(128) vq_main(const float* __restrict__ x,
                                               const float* __restrict__ ws,
                                               float* __restrict__ out) {
    __shared__ int s_idx[4][16];

    const float* enorm  = ws + WS_ENORM;
    const float* embedT = ws + WS_EMBT;          // [NE][DIM]

    const int lane = threadIdx.x & 31;
    const int wid  = threadIdx.x >> 5;
    const int half = lane >> 4;          // 0: lanes 0-15, 1: lanes 16-31
    const int l15  = lane & 15;
    const int row_base = (blockIdx.x * 4 + wid) * 16;

    // ---- load A block: 16 rows x 64 cols, per-lane v2f per k-step --------
    // A 16x4 f32 layout: lanes 0-15 rows M=lane with K={0,1}; lanes 16-31
    // same rows with K={2,3}.  Contiguous float2 in row-major input.
    const float* xrow = x + (size_t)(row_base + l15) * DIM + 2 * half;
    v2f a[KSTEPS];
    #pragma unroll
    for (int t = 0; t < KSTEPS; ++t)
        a[t] = *(const v2f*)(xrow + 4 * t);

    float minv[8];
    int   mini[8];
    #pragma unroll
    for (int v = 0; v < 8; ++v) { minv[v] = 3.4e38f; mini[v] = 0; }

    // ---- sweep 32 code tiles of 16 codes each ----------------------------
    for (int tile = 0; tile < NTILES; ++tile) {
        const int n = tile * 16 + l15;                 // code column this lane
        // B 4x16 f32: lanes = code cols N; VGPR0/1 = K, lanes 16-31 at K+2.
        // embedT is code-major, so each lane reads contiguous float2's.
        const float* bptr = embedT + (size_t)n * DIM + 2 * half;

        v8f acc = {0.f, 0.f, 0.f, 0.f, 0.f, 0.f, 0.f, 0.f};
        #pragma unroll
        for (int t = 0; t < KSTEPS; ++t) {
            v2f b = *(const v2f*)(bptr + 4 * t);
            acc = __builtin_amdgcn_wmma_f32_16x16x4_f32(
                      /*neg_a=*/false, a[t],
                      /*neg_b=*/false, b,
                      /*c_mod=*/(short)0, acc,
                      /*reuse_a=*/false, /*reuse_b=*/false);
        }

        // dist (up to per-row constant) = ||e||^2 - 2 * dot
        const float en = enorm[n];
        #pragma unroll
        for (int v = 0; v < 8; ++v) {
            float d = en - 2.0f * acc[v];
            if (d < minv[v]) { minv[v] = d; mini[v] = n; }
        }
    }

    // ---- argmin across the 16 lanes holding each row's codes -------------
    #pragma unroll
    for (int v = 0; v < 8; ++v) {
        float mv = minv[v];
        int   mi = mini[v];
        #pragma unroll
        for (int off = 8; off >= 1; off >>= 1) {
            float ov = __shfl_xor(mv, off, 32);
            int   oi = __shfl_xor(mi, off, 32);
            if (ov < mv || (ov == mv && oi < mi)) { mv = ov; mi = oi; }
        }
        // C layout: vgpr v = row v (lanes 0-15) / row v+8 (lanes 16-31)
        if (l15 == 0) s_idx[wid][v + (half << 3)] = mi;
    }
    __syncthreads();

    // ---- epilogue: gather codebook rows, store quantize, scatter sums ----
    float dsum = 0.0f;
    float* outq = out + OFF_Q;
    float* es   = out + OFF_ES;
    const int k = lane * 2;   // each lane handles dims k, k+1

    for (int r = 0; r < 16; ++r) {
        const int row = row_base + r;
        const int ci  = s_idx[wid][r];

        // contiguous gather from transposed codebook row
        v2f q = *(const v2f*)(embedT + (size_t)ci * DIM + k);
        float x0 = x[(size_t)row * DIM + k + 0];
        float x1 = x[(size_t)row * DIM + k + 1];

        outq[(size_t)row * DIM + k + 0] = q.x;  // quantize_st == quantize fwd
        outq[(size_t)row * DIM + k + 1] = q.y;

        float d0 = q.x - x0, d1 = q.y - x1;
        dsum += d0 * d0 + d1 * d1;

        atomicAdd(&es[(k + 0) * NE + ci], x0);  // embed_sum[dim, n_embed]
        atomicAdd(&es[(k + 1) * NE + ci], x1);

        if (lane == 0) {
            out[OFF_IND + row] = (float)ci;     // argmin index (as float)
            atomicAdd(&out[OFF_OH + ci], 1.0f); // one-hot count
        }
    }

    // wave-reduce diff partial, single atomic per wave
    #pragma unroll
    for (int off = 16; off >= 1; off >>= 1)
        dsum += __shfl_xor(dsum, off, 32);
    if (lane == 0)
        atomicAdd(&out[OFF_DIFF], dsum * (1.0f / (float)(N_ROWS * DIM)));
}

// ---------------------------------------------------------------------------
extern "C" void kernel_launch(void* const* d_in, const int* in_sizes, int n_in,
                              void* d_out, int out_size, void* d_ws, size_t ws_size,
                              hipStream_t stream) {
    const float* x     = (const float*)d_in[0];   // (32,4096,64) fp32
    const float* embed = (const float*)d_in[1];   // (64,512) fp32
    float* out = (float*)d_out;
    float* ws  = (float*)d_ws;                    // enorm[512] | embedT[512*64]

    vq_init<<<(DIM * NE + 255) / 256, 256, 0, stream>>>(embed, out, ws);
    vq_main<<<N_ROWS / 64, 128, 0, stream>>>(x, ws, out);
}